// MultiModeMultiWavelengthModel_40458591928636
// MI455X (gfx1250) — compile-verified
//
#include <hip/hip_runtime.h>
#include <math.h>

// MI455X / gfx1250: 6-round multi-wavelength angular-spectrum propagation.
// All 512-point FFTs via V_WMMA_F32_16X16X4_F32 (four-step, 512 = 16 x 32),
// planar re/im LDS layouts (every WMMA fragment = one aligned ds_load_b64),
// LDS twiddle table, TDM tensor_load_to_lds line staging for plain row loads,
// async global->LDS staging for fused-pointwise row loads.

#define H      512
#define HH     (512 * 512)
#define NFIELD 72          // 8 batch * 3 modes * 3 wavelengths
#define NWL    3
#define TWO_PI 6.28318530717958647692f

#define AS1 __attribute__((address_space(1)))
#define AS3 __attribute__((address_space(3)))

#if __has_builtin(__builtin_amdgcn_global_load_async_to_lds_b128) && \
    __has_builtin(__builtin_amdgcn_s_wait_asynccnt)
#define USE_ASYNC_LDS 1
#else
#define USE_ASYNC_LDS 0
#endif

#if __has_builtin(__builtin_amdgcn_tensor_load_to_lds) && \
    __has_builtin(__builtin_amdgcn_s_wait_tensorcnt)
#define USE_TDM 1
#else
#define USE_TDM 0
#endif

typedef float v2f __attribute__((ext_vector_type(2)));
typedef float v8f __attribute__((ext_vector_type(8)));
typedef int   v4i __attribute__((vector_size(16)));   // GCC-style, per builtin sig
typedef AS1 v4i* as1_v4i_ptr;                         // pointer to AS1 v4i
typedef AS3 v4i* as3_v4i_ptr;                         // pointer to AS3 v4i
typedef unsigned int u32x4 __attribute__((vector_size(16)));
typedef int i32x8 __attribute__((vector_size(32)));
typedef int i32x4 __attribute__((vector_size(16)));

__device__ __forceinline__ v8f wmma_f32(v2f a, v2f b, v8f c) {
  // D(16x16,f32) += A(16x4,f32) * B(4x16,f32). fp32 WMMA has no A/B negate
  // (ISA NEG = CNeg only), so complex "-Ai*Bi" uses a pre-negated fragment.
  return __builtin_amdgcn_wmma_f32_16x16x4_f32(false, a, false, b, (short)0, c,
                                               false, false);
}

__device__ __forceinline__ float2 cmulf(float2 a, float2 b) {
  return make_float2(a.x * b.x - a.y * b.y, a.x * b.y + a.y * b.x);
}

// ---------------------------------------------------------------------------
// Init kernels.
// ---------------------------------------------------------------------------
__global__ void init_dft_mats(float2* __restrict__ w16f, float2* __restrict__ w16i,
                              float2* __restrict__ w32f, float2* __restrict__ w32i) {
  for (int i = threadIdx.x; i < 256; i += blockDim.x) {
    int r = i >> 4, c = i & 15;
    float ang = -TWO_PI * (float)(r * c) / 16.0f;
    float s_, c_;
    __sincosf(ang, &s_, &c_);
    w16f[i] = make_float2(c_, s_);
    w16i[i] = make_float2(c_, -s_);
  }
  for (int i = threadIdx.x; i < 1024; i += blockDim.x) {
    int r = i >> 5, c = i & 31;
    float ang = -TWO_PI * (float)(r * c) / 32.0f;
    float s_, c_;
    __sincosf(ang, &s_, &c_);
    w32f[i] = make_float2(c_, s_);
    w32i[i] = make_float2(c_, -s_);
  }
}

__global__ void init_prop(float2* __restrict__ prop, float z) {
  int idx = blockIdx.x * blockDim.x + threadIdx.x;
  if (idx >= NWL * HH) return;
  const float wls[NWL] = {5.32e-07f, 6.33e-07f, 8.5e-07f};
  int w   = idx / HH;
  int rem = idx - w * HH;
  int i   = rem >> 9;
  int j   = rem & 511;
  float fi = (float)((i < 256) ? i : i - 512) / (512.0f * 8e-06f);
  float fj = (float)((j < 256) ? j : j - 512) / (512.0f * 8e-06f);
  float kx = TWO_PI * fi;
  float ky = TWO_PI * fj;
  float k  = TWO_PI / wls[w];
  float arg = k * k - kx * kx - ky * ky;
  float2 v;
  if (arg >= 0.0f) {
    float kz = sqrtf(arg);
    float s_, c_;
    __sincosf(kz * z, &s_, &c_);
    v = make_float2(c_, s_);
  } else {
    v = make_float2(__expf(-sqrtf(-arg) * z), 0.0f);
  }
  prop[idx] = v;
}

// ---------------------------------------------------------------------------
// One 1-D transform pass. Each wave owns one 512-complex line at a time.
//   mode 0: plain load           mode 1: load (x_real,x_imag)*exp(i*phase)
//   mode 2: load field*exp(i*ph) mode 3: load field*prop_kernel
//   axis 0: rows, axis 1: columns. twsign -1 fwd / +1 inv; scale folds 1/N.
// LDS layouts (planar re/im so WMMA fragments are single b64 loads):
//   rb:  X[n1][n2]  re at [n1 + 16*n2],       im at +512
//   yb:  Y'[k1][n2] re at [n2 + 32*k1],       im at +512  (also DMA staging)
//   w2:  W32^T      re at [n*32 + k],         im separate
//   tw:  twiddle    cos at [k1 + 16*n2],      sin separate
// ---------------------------------------------------------------------------
__global__ __launch_bounds__(128) void fft512_pass(
    const float2* __restrict__ bufIn, float2* __restrict__ bufOut,
    const float* __restrict__ xr, const float* __restrict__ xi,
    const float* __restrict__ ph, const float2* __restrict__ prop,
    const float2* __restrict__ w1, const float2* __restrict__ w2,
    int axis, int mode, float twsign, float scale) {
  __shared__ float2 w1buf[256];   // W16, row-major interleaved (A-frag preload)
  __shared__ float  w2re[1024];   // W32 transposed, planar
  __shared__ float  w2im[1024];
  __shared__ float  twre[512];    // twiddle table
  __shared__ float  twim[512];
  __shared__ float  rb[4][1024];  // per-wave line, planar
  __shared__ float  yb[4][1024];  // per-wave stage-1 result, planar (+staging)

  const int tid  = threadIdx.x;
  const int lane = tid & 31;
  const int wave = tid >> 5;
  const int field = blockIdx.x;
  const int wl = field % NWL;
  const size_t fbase = (size_t)field * HH;
  const float* phw = ph + (size_t)wl * HH;
  const float2* prw = prop + (size_t)wl * HH;

  for (int i = tid; i < 256; i += 128) w1buf[i] = w1[i];
  for (int i = tid; i < 1024; i += 128) {
    int r = i >> 5, c = i & 31;
    float2 v = w2[i];
    w2re[c * 32 + r] = v.x;     // transpose: [n*32 + k]
    w2im[c * 32 + r] = v.y;
  }
  for (int i = tid; i < 512; i += 128) {
    int k1 = i & 15, n2 = i >> 4;
    float ang = twsign * (TWO_PI / 512.0f) * (float)(k1 * n2);
    float s_, c_;
    __sincosf(ang, &s_, &c_);
    twre[i] = c_;
    twim[i] = s_;
  }
  __syncthreads();

  const int m    = lane & 15;          // A-row / B-col / C-col index
  const int koff = (lane >> 4) << 1;   // K sub-offset: 0 or 2
  const int hi   = (lane >> 4) << 3;   // C/D row offset: 0 or 8

  // Preload W16 A-fragments (4 chained K=4 chunks; re, im, -im).
  v2f a1r[4], a1i[4], a1n[4];
#pragma unroll
  for (int c = 0; c < 4; ++c) {
    int k0 = 4 * c + koff;
    float2 wa = w1buf[m * 16 + k0];
    float2 wb = w1buf[m * 16 + k0 + 1];
    a1r[c] = (v2f){wa.x, wb.x};
    a1i[c] = (v2f){wa.y, wb.y};
    a1n[c] = -a1i[c];
  }

  float* rbre = rb[wave];
  float* rbim = rbre + 512;
  float* ybre = yb[wave];
  float* ybim = ybre + 512;
  float2* sg = (float2*)yb[wave];      // 4KB staging alias (disjoint lifetime)

  for (int it = 0; it < 16; ++it) {
    const int line = blockIdx.y * 64 + wave * 16 + it;

    if (mode != 1 && line + 1 < H) {
      size_t nidx = fbase + ((axis == 0) ? (size_t)(line + 1) * H + lane * 16
                                         : (size_t)(lane * 16) * H + (line + 1));
      __builtin_prefetch((const void*)(bufIn + nidx), 0, 1);
    }

    // ---- gather one transform line into planar LDS, fused pointwise op ----
#if USE_TDM
    if (axis == 0 && mode == 0) {
      // One TDM descriptor DMAs the whole contiguous 4KB line into LDS:
      // 1-D tile, data_size=8B, tile_dim0 = tensor_dim0 = 512.
      unsigned ldsa =
          __builtin_amdgcn_readfirstlane((unsigned)(size_t)(void*)sg);
      unsigned long long ga =
          (unsigned long long)(size_t)(const void*)(bufIn + fbase +
                                                    (size_t)line * H);
      unsigned galo = __builtin_amdgcn_readfirstlane((unsigned)ga);
      unsigned gahi = __builtin_amdgcn_readfirstlane((unsigned)(ga >> 32));
      u32x4 g0 = {1u,                     // count=1, user descriptor
                  ldsa,                   // lds_addr
                  galo,                   // global_addr[31:0]
                  (gahi & 0x01FFFFFFu) | (2u << 30)};  // ga[56:32] | type=2
      i32x8 g1 = {(int)(3u << 16),        // data_size=8B (mask=0, no pad)
                  (int)(512u << 16),      // tensor_dim0[15:0]=512 (bits 63:48)
                  (int)(1u << 16),        // tensor_dim1[15:0]=1
                  (int)(512u << 16),      // tile_dim0=512 (bits 127:112)
                  0,                      // tile_dim1=0 (unused), tile_dim2=0
                  512, 0,                 // tensor_dim0_stride=512
                  0};                     // tensor_dim1_stride=0
      i32x4 gz4 = {0, 0, 0, 0};
      i32x8 gz8 = {0, 0, 0, 0, 0, 0, 0, 0};
      __builtin_amdgcn_tensor_load_to_lds(g0, g1, gz4, gz4, gz8, 0);
      __builtin_amdgcn_s_wait_tensorcnt(0);
      for (int e = lane; e < H; e += 32) {
        float2 v = sg[e];
        rbre[e] = v.x;
        rbim[e] = v.y;
      }
    } else
#endif
#if USE_ASYNC_LDS
    if (axis == 0 && mode != 1) {
      // Row lines are contiguous: async-DMA 4KB into LDS staging (8 x b128).
      const float2* gsrc = bufIn + fbase + (size_t)line * H;
      char* sgc = (char*)sg;
#pragma unroll
      for (int cp = 0; cp < 8; ++cp) {
        const int ch = lane + 32 * cp;   // 16-byte chunk id, 0..255
        __builtin_amdgcn_global_load_async_to_lds_b128(
            (as1_v4i_ptr)(AS1 void*)(void*)(gsrc + 2 * ch),
            (as3_v4i_ptr)(AS3 void*)(void*)(sgc + 16 * ch), 0, 0);
      }
      __builtin_amdgcn_s_wait_asynccnt(0);
      for (int e = lane; e < H; e += 32) {
        float2 v = sg[e];
        if (mode == 2) {
          float s_, c_;
          __sincosf(phw[(size_t)line * H + e], &s_, &c_);
          v = cmulf(v, make_float2(c_, s_));
        } else if (mode == 3) {
          v = cmulf(v, prw[(size_t)line * H + e]);
        }
        rbre[e] = v.x;
        rbim[e] = v.y;
      }
    } else
#endif
    {
      for (int e = lane; e < H; e += 32) {
        const size_t lidx = (axis == 0) ? (size_t)line * H + e
                                        : (size_t)e * H + line;
        const size_t gidx = fbase + lidx;
        float2 v;
        if (mode == 1) {
          float s_, c_;
          __sincosf(phw[lidx], &s_, &c_);
          float re = xr[gidx], im = xi[gidx];
          v = make_float2(re * c_ - im * s_, re * s_ + im * c_);
        } else if (mode == 2) {
          float s_, c_;
          __sincosf(phw[lidx], &s_, &c_);
          v = cmulf(bufIn[gidx], make_float2(c_, s_));
        } else if (mode == 3) {
          v = cmulf(bufIn[gidx], prw[lidx]);
        } else {
          v = bufIn[gidx];
        }
        rbre[e] = v.x;
        rbim[e] = v.y;
      }
    }
    __syncthreads();

    // ---- stage 1: Y(16x32) = W16 * X, then twiddle (from LDS table) ----
#pragma unroll
    for (int t = 0; t < 2; ++t) {
      v8f cre = {};
      v8f cim = {};
      const int n2 = 16 * t + m;
      const float* brr = rbre + 16 * n2;
      const float* bri = rbim + 16 * n2;
#pragma unroll
      for (int c = 0; c < 4; ++c) {
        const int k0 = 4 * c + koff;          // even -> 8B-aligned b64
        v2f br = *(const v2f*)(brr + k0);
        v2f bi = *(const v2f*)(bri + k0);
        cre = wmma_f32(a1r[c], br, cre);
        cre = wmma_f32(a1n[c], bi, cre);
        cim = wmma_f32(a1r[c], bi, cim);
        cim = wmma_f32(a1i[c], br, cim);
      }
#pragma unroll
      for (int r = 0; r < 8; ++r) {
        const int k1 = r + hi;
        const float tc = twre[k1 + 16 * n2];
        const float ts = twim[k1 + 16 * n2];
        ybre[n2 + 32 * k1] = cre[r] * tc - cim[r] * ts;
        ybim[n2 + 32 * k1] = cre[r] * ts + cim[r] * tc;
      }
    }
    __syncthreads();

    // ---- stage 2: Z(16x32) = Y' * W32 (transposed planar in LDS) ----
#pragma unroll
    for (int t2 = 0; t2 < 2; ++t2) {
      v8f dre = {};
      v8f dim_ = {};
      const int nn = 16 * t2 + m;
      const float* wbr = w2re + nn * 32;
      const float* wbi = w2im + nn * 32;
      const float* yar = ybre + 32 * m;
      const float* yai = ybim + 32 * m;
#pragma unroll
      for (int c = 0; c < 8; ++c) {
        const int kk = 4 * c + koff;          // even -> aligned b64
        v2f ar = *(const v2f*)(yar + kk);
        v2f ai = *(const v2f*)(yai + kk);
        v2f an = -ai;
        v2f br = *(const v2f*)(wbr + kk);
        v2f bi = *(const v2f*)(wbi + kk);
        dre = wmma_f32(ar, br, dre);
        dre = wmma_f32(an, bi, dre);
        dim_ = wmma_f32(ar, bi, dim_);
        dim_ = wmma_f32(ai, br, dim_);
      }
#pragma unroll
      for (int r = 0; r < 8; ++r) {
        const int k1 = r + hi;
        const int outj = k1 * 32 + nn;
        const size_t oidx = fbase + ((axis == 0) ? (size_t)line * H + outj
                                                 : (size_t)outj * H + line);
        bufOut[oidx] = make_float2(dre[r] * scale, dim_[r] * scale);
      }
    }
    __syncthreads();
  }
}

// ---------------------------------------------------------------------------
// 6 rounds x (rowF+mod, colF, colI*prop, rowI) = 24 passes; ping-pong between
// d_ws field buffer and d_out so pass 24 lands in d_out.
// ---------------------------------------------------------------------------
extern "C" void kernel_launch(void* const* d_in, const int* in_sizes, int n_in,
                              void* d_out, int out_size, void* d_ws,
                              size_t ws_size, hipStream_t stream) {
  (void)in_sizes; (void)n_in; (void)out_size; (void)ws_size;
  const float* xr = (const float*)d_in[0];
  const float* xi = (const float*)d_in[1];
  const float* pm = (const float*)d_in[2];  // [5][3][H][H]

  float2* fieldA = (float2*)d_ws;                       // 72*HH complex
  float2* propL  = fieldA + (size_t)NFIELD * HH;        // 3*HH
  float2* propD  = propL + (size_t)NWL * HH;            // 3*HH
  float2* w16f   = propD + (size_t)NWL * HH;            // 256
  float2* w16i   = w16f + 256;                          // 256
  float2* w32f   = w16i + 256;                          // 1024
  float2* w32i   = w32f + 1024;                         // 1024

  init_dft_mats<<<1, 256, 0, stream>>>(w16f, w16i, w32f, w32i);
  init_prop<<<(NWL * HH + 255) / 256, 256, 0, stream>>>(propL, 0.03f);
  init_prop<<<(NWL * HH + 255) / 256, 256, 0, stream>>>(propD, 0.1f);

  float2* bufs[2] = {fieldA, (float2*)d_out};
  dim3 grid(NFIELD, 8), block(128);
  const float FWD = -1.0f, INV = 1.0f, INVSCALE = 1.0f / 512.0f;

  int dst = 0;
  const float2* src = nullptr;
  for (int l = 0; l < 6; ++l) {
    const bool det = (l == 5);
    const float* phl = det ? pm : (pm + (size_t)l * NWL * HH);
    const float2* pk = det ? propD : propL;
    const int mod = det ? 0 : (l == 0 ? 1 : 2);
    // row forward (fused input modulation)
    fft512_pass<<<grid, block, 0, stream>>>(src, bufs[dst], xr, xi, phl, pk,
                                            w16f, w32f, 0, mod, FWD, 1.0f);
    src = bufs[dst]; dst ^= 1;
    // column forward
    fft512_pass<<<grid, block, 0, stream>>>(src, bufs[dst], xr, xi, phl, pk,
                                            w16f, w32f, 1, 0, FWD, 1.0f);
    src = bufs[dst]; dst ^= 1;
    // column inverse with fused propagation-kernel multiply
    fft512_pass<<<grid, block, 0, stream>>>(src, bufs[dst], xr, xi, phl, pk,
                                            w16i, w32i, 1, 3, INV, INVSCALE);
    src = bufs[dst]; dst ^= 1;
    // row inverse (lands in d_out on the final round)
    fft512_pass<<<grid, block, 0, stream>>>(src, bufs[dst], xr, xi, phl, pk,
                                            w16i, w32i, 0, 0, INV, INVSCALE);
    src = bufs[dst]; dst ^= 1;
  }
}